// MemOps_10161892622458
// MI455X (gfx1250) — compile-verified
//
#include <hip/hip_runtime.h>
#include <math.h>

typedef __attribute__((ext_vector_type(2))) float v2f;
typedef __attribute__((ext_vector_type(4))) float v4f;
typedef __attribute__((ext_vector_type(8))) float v8f;

#define BSZ     16
#define KNEG    65536
#define KTOT    65537          // K + 1 (positive + negatives)
#define NDIM    128
#define NDATA   1000000
#define INV_T   (1.0f / 0.07f)
#define TILES_PER_B 4097       // ceil(65537 / 16)
#define LDS_STRIDE  132        // 128 + 4 pad -> conflict-free A-fragment reads

// ---------------------------------------------------------------------------
// Kernel 0: zero the z0 accumulator (d_ws is poisoned, not re-poisoned)
// ---------------------------------------------------------------------------
__global__ void init_accum_kernel(float* accum) {
    if (threadIdx.x == 0 && blockIdx.x == 0) accum[0] = 0.0f;
}

// ---------------------------------------------------------------------------
// Kernel 1: gathered batched GEMV via V_WMMA_F32_16X16X4_F32.
// One wave (32 threads) per (sample b, 16-row tile). LDS-staged gather.
// B matrix: column 0 = x_b, column 1 = z_b, columns 2..15 = 0 (zero LDS
// region selected branch-free per lane -> no EXEC churn in the WMMA loop).
// Writes UNSCALED exp(logit) for lx and lz; accumulates sum(exp(lx)).
// ---------------------------------------------------------------------------
__global__ __launch_bounds__(32)
void logits_wmma_kernel(const float* __restrict__ x,
                        const float* __restrict__ z,
                        const float* __restrict__ mem,
                        const int*   __restrict__ y,
                        const int*   __restrict__ idx,
                        float*       __restrict__ out,   // [2][BSZ][KTOT] region
                        float*       __restrict__ accum)
{
    __shared__ float ldsA[16 * LDS_STRIDE];
    __shared__ float ldsB[3 * NDIM];     // [x | z | zeros]
    __shared__ int   rows[16];

    const int blk  = blockIdx.x;
    const int b    = blk / TILES_PER_B;
    const int tile = blk % TILES_PER_B;
    const int lane = threadIdx.x;

    // Row indices for this 16-row tile: k==0 -> positive row y[b], else idx.
    if (lane < 16) {
        int kg = tile * 16 + lane;
        int r  = 0;
        if (kg == 0)          r = y[b];
        else if (kg <= KNEG)  r = idx[b * KNEG + (kg - 1)];
        rows[lane] = r;                  // kg > KNEG: dummy row 0 (masked later)
    }
    // Stage x_b, z_b and a zero region (128 floats each; 32 lanes x float4).
    {
        v4f xv = *(const v4f*)(x + b * NDIM + lane * 4);
        v4f zv = *(const v4f*)(z + b * NDIM + lane * 4);
        v4f zero = {};
        *(v4f*)(ldsB + 0 * NDIM + lane * 4) = xv;
        *(v4f*)(ldsB + 1 * NDIM + lane * 4) = zv;
        *(v4f*)(ldsB + 2 * NDIM + lane * 4) = zero;
    }
    __syncthreads();

    // Coalesced gather: iteration i loads row i (512B) across the wave.
    for (int i = 0; i < 16; ++i) {
        const float* src = mem + (size_t)rows[i] * NDIM + lane * 4;
        v4f v = *(const v4f*)src;
        *(v4f*)(ldsA + i * LDS_STRIDE + lane * 4) = v;
    }
    __syncthreads();

    const int m16  = lane & 15;
    const int half = lane >> 4;          // 0: K pair {0,1}; 1: K pair {2,3}

    // Per-lane fragment base pointers (selected ONCE, branch-free).
    const int sel = (m16 < 2) ? m16 : 2;                   // 0=x, 1=z, 2=zero
    const float* aptr = ldsA + m16 * LDS_STRIDE + half * 2;
    const float* bptr = ldsB + sel * NDIM      + half * 2;

    // B is loop-invariant per wave: preload all 32 fragments into registers.
    v2f breg[32];
    #pragma unroll
    for (int i = 0; i < 32; ++i)
        breg[i] = *(const v2f*)(bptr + i * 4);

    v8f c = {};
    #pragma unroll
    for (int i = 0; i < 32; ++i) {
        v2f a = *(const v2f*)(aptr + i * 4);   // A: 16x4 f32 fragment
        c = __builtin_amdgcn_wmma_f32_16x16x4_f32(
                false, a, false, breg[i], (short)0, c, false, false);
    }

    // D layout: lane N = m16; VGPR r -> M = r + 8*half.
    // Column 0 = lx dots, column 1 = lz dots.
    if (m16 < 2) {
        float* dst = out + (m16 == 0 ? (size_t)0 : (size_t)BSZ * KTOT);
        const int mbase = half * 8;
        float psum = 0.0f;
        #pragma unroll
        for (int r = 0; r < 8; ++r) {
            int kg = tile * 16 + mbase + r;
            if (kg < KTOT) {
                float e = expf(c[r] * INV_T);
                dst[(size_t)b * KTOT + kg] = e;
                if (m16 == 0) psum += e;
            }
        }
        if (m16 == 0) atomicAdd(accum, psum);
    }
}

// ---------------------------------------------------------------------------
// Kernel 2: scale lx and lz by 1/z0, z0 = mean(exp(lx)) * NDATA
// ---------------------------------------------------------------------------
__global__ void scale_kernel(float* __restrict__ out,
                             const float* __restrict__ accum)
{
    const float z0 = accum[0] * ((float)NDATA / (float)(BSZ * KTOT));
    const float s  = 1.0f / z0;
    const size_t n = (size_t)2 * BSZ * KTOT;
    for (size_t i = (size_t)blockIdx.x * blockDim.x + threadIdx.x; i < n;
         i += (size_t)gridDim.x * blockDim.x)
        out[i] *= s;
}

// ---------------------------------------------------------------------------
// Kernel 3: bulk copy memory -> new_memory output region (float4 streams)
// ---------------------------------------------------------------------------
__global__ void copy_mem_kernel(const v4f* __restrict__ src,
                                v4f* __restrict__ dst, size_t n4)
{
    for (size_t i = (size_t)blockIdx.x * blockDim.x + threadIdx.x; i < n4;
         i += (size_t)gridDim.x * blockDim.x)
        dst[i] = src[i];
}

// ---------------------------------------------------------------------------
// Kernel 4: EMA scatter update of 16 rows (sequential b -> last-dup wins).
// Reads ORIGINAL memory (matches reference m_pos semantics).
// ---------------------------------------------------------------------------
__global__ __launch_bounds__(NDIM)
void update_rows_kernel(const float* __restrict__ x,
                        const float* __restrict__ mem,
                        const int*   __restrict__ y,
                        float*       __restrict__ out_mem)
{
    __shared__ float red[NDIM];
    const int t = threadIdx.x;
    for (int b = 0; b < BSZ; ++b) {
        const int row = y[b];
        const float w = 0.5f * mem[(size_t)row * NDIM + t]
                      + 0.5f * x[b * NDIM + t];
        red[t] = w * w;
        __syncthreads();
        #pragma unroll
        for (int s = NDIM / 2; s > 0; s >>= 1) {
            if (t < s) red[t] += red[t + s];
            __syncthreads();
        }
        const float inv = 1.0f / sqrtf(red[0]);
        __syncthreads();   // protect red[] before next iteration's writes
        out_mem[(size_t)row * NDIM + t] = w * inv;
    }
}

// ---------------------------------------------------------------------------
extern "C" void kernel_launch(void* const* d_in, const int* in_sizes, int n_in,
                              void* d_out, int out_size, void* d_ws, size_t ws_size,
                              hipStream_t stream)
{
    const float* x   = (const float*)d_in[0];   // [16, 128]
    const float* z   = (const float*)d_in[1];   // [16, 128]
    const float* mem = (const float*)d_in[2];   // [1e6, 128]
    const int*   y   = (const int*)d_in[3];     // [16]
    const int*   idx = (const int*)d_in[4];     // [16, 65536]

    float* out     = (float*)d_out;                          // lx | lz | memory
    float* out_mem = out + (size_t)2 * BSZ * KTOT;
    float* accum   = (float*)d_ws;

    init_accum_kernel<<<1, 1, 0, stream>>>(accum);

    logits_wmma_kernel<<<BSZ * TILES_PER_B, 32, 0, stream>>>(
        x, z, mem, y, idx, out, accum);

    scale_kernel<<<2048, 256, 0, stream>>>(out, accum);

    const size_t n4 = (size_t)NDATA * NDIM / 4;              // 32M float4
    copy_mem_kernel<<<8192, 256, 0, stream>>>(
        (const v4f*)mem, (v4f*)out_mem, n4);

    update_rows_kernel<<<1, NDIM, 0, stream>>>(x, mem, y, out_mem);
}